// CoarseGraining_53798760350209
// MI455X (gfx1250) — compile-verified
//
#include <hip/hip_runtime.h>
#include <hip/hip_bf16.h>
#include <math.h>

#define NPTS 8192
#define MPTS 8192
#define NB   16
#define WIDTH 32
#define NSPLIT 8
#define CHUNK 128
#define PI_F 3.14159265358979323846f
#define LOG2_F 0.69314718055994530942f
#define LOG2E_F 1.44269504088896340736f

typedef float v2f __attribute__((ext_vector_type(2)));
typedef float v8f __attribute__((ext_vector_type(8)));

// ---------------------------------------------------------------------------
// Phase 1: per-source prep.  negb[j][b] = -beta[j][b] * log2(e), wrho, |c_j|^2
// ---------------------------------------------------------------------------
__global__ __launch_bounds__(256) void source_kernel(
    const float* __restrict__ rho, const float* __restrict__ gamma,
    const float* __restrict__ coords, const float* __restrict__ weights,
    const float* __restrict__ w1, const float* __restrict__ b1,
    const float* __restrict__ w2, const float* __restrict__ b2,
    float* __restrict__ negb, float* __restrict__ wrho, float* __restrict__ cnorm)
{
    int j = blockIdx.x * 256 + threadIdx.x;
    if (j >= NPTS) return;
    float r = rho[j];
    float g = gamma[j];
    const float kf = 4.0f * powf(3.0f * PI_F * PI_F, 2.0f / 3.0f);
    float s2 = g / (kf * powf(r, 8.0f / 3.0f));
    float x = logf(s2 + 1e-4f);

    float e[NB];
#pragma unroll
    for (int b = 0; b < NB; ++b) e[b] = b2[b];
#pragma unroll 4
    for (int k = 0; k < WIDTH; ++k) {
        float h = tanhf(fmaf(x, w1[k], b1[k]));
#pragma unroll
        for (int b = 0; b < NB; ++b) e[b] = fmaf(h, w2[k * NB + b], e[b]);
    }
    float pref = PI_F * powf(0.5f * r, 2.0f / 3.0f);
#pragma unroll
    for (int b = 0; b < NB; ++b) {
        float a = fabsf(e[b]);
        float lc = a + log1pf(expf(-2.0f * a)) - LOG2_F;   // log(cosh(e))
        negb[j * NB + b] = -pref * lc * LOG2E_F;
    }
    wrho[j] = weights[j] * r;
    float cx = coords[3 * j + 0], cy = coords[3 * j + 1], cz = coords[3 * j + 2];
    cnorm[j] = cx * cx + cy * cy + cz * cz;
}

// ---------------------------------------------------------------------------
// HEG scale: log_cosh(MLP(0))^1.5  (16 values)
// ---------------------------------------------------------------------------
__global__ void heg_kernel(const float* __restrict__ b1, const float* __restrict__ w2,
                           const float* __restrict__ b2, float* __restrict__ heg)
{
    int b = threadIdx.x;
    if (b >= NB) return;
    float e = b2[b];
    for (int k = 0; k < WIDTH; ++k)
        e = fmaf(tanhf(b1[k]), w2[k * NB + b], e);
    float a = fabsf(e);
    float lc = a + log1pf(expf(-2.0f * a)) - LOG2_F;
    heg[b] = powf(lc, 1.5f);
}

// ---------------------------------------------------------------------------
// Phase 2: main pairwise kernel.
// Wave handles 16 output points (m = WMMA N-dim), all 16 bases.
// d2 tile (16 j x 16 m) via V_WMMA_F32_16X16X4_F32:
//   A[j,:] = (cx, cy, cz, |c_j|^2)
//   B[:,m] = (-2ox, -2oy, -2oz, 1)           (loop-invariant per wave)
//   C      = 0 ; |o_m|^2 added per-lane after (fuses with the clamp)
// Lane layout of D: half h = lane/16 holds rows j = v + 8h (v = VGPR 0..7),
// column m = lane%16.  Each lane accumulates y[m, 0..15] over its j's;
// lane L and L+16 partials combined once at the end via shfl_xor(16).
// ---------------------------------------------------------------------------
__global__ __launch_bounds__(256) void pair_kernel(
    const float* __restrict__ coords, const float* __restrict__ out_coords,
    const float* __restrict__ negb, const float* __restrict__ wrho,
    const float* __restrict__ cnorm, float* __restrict__ partial)
{
    __shared__ float sNB[CHUNK * NB];
    __shared__ float sW[CHUNK];
    __shared__ float sCN[CHUNK];
    __shared__ float sAX[CHUNK];
    __shared__ float sAY[CHUNK];
    __shared__ float sAZ[CHUNK];

    const int tid  = threadIdx.x;
    const int lane = tid & 31;
    const int wave = tid >> 5;
    const int half = lane >> 4;      // 0: rows j=v, K=0..1 | 1: rows j=v+8, K=2..3
    const int lm   = lane & 15;      // output column within tile
    const int m    = blockIdx.x * 128 + wave * 16 + lm;
    const int jbase = blockIdx.y * (NPTS / NSPLIT);

    // Loop-invariant B matrix
    float ox = out_coords[m * 3 + 0];
    float oy = out_coords[m * 3 + 1];
    float oz = out_coords[m * 3 + 2];
    float onorm = ox * ox + oy * oy + oz * oz;
    v2f bmat;
    bmat.x = half ? (-2.0f * oz) : (-2.0f * ox);
    bmat.y = half ? 1.0f         : (-2.0f * oy);

    float acc[NB];
#pragma unroll
    for (int b = 0; b < NB; ++b) acc[b] = 0.0f;

    for (int c0 = 0; c0 < NPTS / NSPLIT; c0 += CHUNK) {
        __syncthreads();   // protect LDS from previous iteration's readers
        const int jg = jbase + c0;
        // Stage chunk into LDS (shared by all 8 waves of the block)
        for (int t = tid; t < CHUNK * NB / 4; t += 256)
            ((float4*)sNB)[t] = ((const float4*)(negb + (size_t)jg * NB))[t];
        for (int t = tid; t < CHUNK; t += 256) {
            sW[t]  = wrho[jg + t];
            sCN[t] = cnorm[jg + t];
        }
        // coords: contiguous global reads, scattered SoA writes to LDS
        for (int t = tid; t < CHUNK * 3; t += 256) {
            float v = coords[(size_t)jg * 3 + t];
            int jj = t / 3, k = t - 3 * jj;
            if (k == 0)      sAX[jj] = v;
            else if (k == 1) sAY[jj] = v;
            else             sAZ[jj] = v;
        }
        __syncthreads();

        for (int jt = 0; jt < CHUNK; jt += 16) {
            // A matrix: half 0 -> (cx, cy); half 1 -> (cz, |c_j|^2)
            int ja = jt + lm;
            v2f amat;
            amat.x = half ? sAZ[ja] : sAX[ja];
            amat.y = half ? sCN[ja] : sAY[ja];

            v8f czero = {};
            v8f d2 = __builtin_amdgcn_wmma_f32_16x16x4_f32(
                false, amat, false, bmat, (short)0, czero, false, false);

#pragma unroll
            for (int v = 0; v < 8; ++v) {
                float dd = fmaxf(d2[v] + onorm, 0.0f);   // v_add + v_max_num
                int jj = jt + v + 8 * half;
                float w = sW[jj];
                const float4* nb4 = (const float4*)(sNB + jj * NB);
#pragma unroll
                for (int q = 0; q < 4; ++q) {
                    float4 nb = nb4[q];   // lane-half-uniform LDS broadcast
                    acc[q * 4 + 0] = fmaf(__builtin_amdgcn_exp2f(nb.x * dd), w, acc[q * 4 + 0]);
                    acc[q * 4 + 1] = fmaf(__builtin_amdgcn_exp2f(nb.y * dd), w, acc[q * 4 + 1]);
                    acc[q * 4 + 2] = fmaf(__builtin_amdgcn_exp2f(nb.z * dd), w, acc[q * 4 + 2]);
                    acc[q * 4 + 3] = fmaf(__builtin_amdgcn_exp2f(nb.w * dd), w, acc[q * 4 + 3]);
                }
            }
        }
    }

    // Combine the two j-halves (lane L and L+16 share output point m)
#pragma unroll
    for (int b = 0; b < NB; ++b) acc[b] += __shfl_xor(acc[b], 16, 32);

    if (half == 0) {
        float* dst = partial + ((size_t)blockIdx.y * MPTS + m) * NB;
#pragma unroll
        for (int q = 0; q < 4; ++q) {
            float4 o;
            o.x = acc[q * 4 + 0]; o.y = acc[q * 4 + 1];
            o.z = acc[q * 4 + 2]; o.w = acc[q * 4 + 3];
            ((float4*)dst)[q] = o;
        }
    }
}

// ---------------------------------------------------------------------------
// Phase 3: reduce splits, apply HEG scale
// ---------------------------------------------------------------------------
__global__ __launch_bounds__(256) void reduce_kernel(
    const float* __restrict__ partial, const float* __restrict__ heg,
    float* __restrict__ out)
{
    int idx = blockIdx.x * 256 + threadIdx.x;   // m*NB + b
    if (idx >= MPTS * NB) return;
    int b = idx & (NB - 1);
    float s = 0.0f;
#pragma unroll
    for (int sp = 0; sp < NSPLIT; ++sp)
        s += partial[(size_t)sp * MPTS * NB + idx];
    out[idx] = s * heg[b];
}

// ---------------------------------------------------------------------------
extern "C" void kernel_launch(void* const* d_in, const int* in_sizes, int n_in,
                              void* d_out, int out_size, void* d_ws, size_t ws_size,
                              hipStream_t stream)
{
    const float* rho        = (const float*)d_in[0];
    const float* gamma      = (const float*)d_in[1];
    const float* coords     = (const float*)d_in[2];
    const float* weights    = (const float*)d_in[3];
    const float* out_coords = (const float*)d_in[4];
    const float* w1         = (const float*)d_in[5];
    const float* b1         = (const float*)d_in[6];
    const float* w2         = (const float*)d_in[7];
    const float* b2         = (const float*)d_in[8];
    float* out = (float*)d_out;

    float* ws      = (float*)d_ws;
    float* negb    = ws;                         // N*NB
    float* wrho    = negb + (size_t)NPTS * NB;   // N
    float* cnorm   = wrho + NPTS;                // N
    float* heg     = cnorm + NPTS;               // NB
    float* partial = heg + NB;                   // NSPLIT*M*NB

    source_kernel<<<NPTS / 256, 256, 0, stream>>>(rho, gamma, coords, weights,
                                                  w1, b1, w2, b2,
                                                  negb, wrho, cnorm);
    heg_kernel<<<1, NB, 0, stream>>>(b1, w2, b2, heg);
    pair_kernel<<<dim3(MPTS / 128, NSPLIT), 256, 0, stream>>>(
        coords, out_coords, negb, wrho, cnorm, partial);
    reduce_kernel<<<(MPTS * NB + 255) / 256, 256, 0, stream>>>(partial, heg, out);
}